// MaskedMHA_46677704573128
// MI455X (gfx1250) — compile-verified
//
#include <hip/hip_runtime.h>

typedef float v2f __attribute__((ext_vector_type(2)));
typedef float v8f __attribute__((ext_vector_type(8)));

#define EMBD 1024
#define HEADS 16
#define DH 64
#define WIN 257
#define TSEQ 4096
#define BSZ 4

__device__ __forceinline__ v8f wmma_f32(v2f a, v2f b, v8f c) {
    // D = A(16x4 f32) * B(4x16 f32) + C(16x16 f32)
    return __builtin_amdgcn_wmma_f32_16x16x4_f32(false, a, false, b, (short)0, c,
                                                 false, false);
}

__device__ __forceinline__ v8f v8f_zero() {
    v8f z = {0.f, 0.f, 0.f, 0.f, 0.f, 0.f, 0.f, 0.f};
    return z;
}

// -------------------------------------------------------------------------
// Y[b,m,n] = sum_k W[m,k] * X[b,k,n] + bias[m]     (1x1 conv == GEMM)
// Workgroup: 256 threads (8 waves). Tile: 128M x 128N, K-panel 16.
// -------------------------------------------------------------------------
#define GPW 17    // Wt row pad (floats)  -> conflict-free A reads
#define GPX 136   // Xt row pad (floats)  -> conflict-free B reads
__global__ __launch_bounds__(256)
void gemm1x1_bias(const float* __restrict__ W, const float* __restrict__ bias,
                  const float* __restrict__ X, float* __restrict__ Y,
                  int M, int K, int N) {
    __shared__ float Wt[128 * GPW];
    __shared__ float Xt[16 * GPX];

    const int tid = threadIdx.x;
    const int wave = tid >> 5;
    const int lane = tid & 31;
    const int Lm = lane & 15;
    const int hi = lane >> 4;
    const int off = hi * 2;
    const int m0 = blockIdx.y * 128;
    const int n0 = blockIdx.x * 128;
    const int b = blockIdx.z;

    const float* Xb = X + (size_t)b * K * N;
    float* Yb = Y + (size_t)b * M * N;

    v8f acc[8];
#pragma unroll
    for (int i = 0; i < 8; ++i) acc[i] = v8f_zero();

    const int wr = tid >> 1;          // W loader: row 0..127
    const int wk = (tid & 1) * 8;     // 8 contiguous k
    const int xr = tid >> 4;          // X loader: k-row 0..15
    const int xc = (tid & 15) * 8;    // 8 contiguous n

    for (int k0 = 0; k0 < K; k0 += 16) {
        __syncthreads();
        const float* wsrc = W + (size_t)(m0 + wr) * K + k0 + wk;
#pragma unroll
        for (int u = 0; u < 8; ++u) Wt[wr * GPW + wk + u] = wsrc[u];
        const float* xsrc = Xb + (size_t)(k0 + xr) * N + n0 + xc;
#pragma unroll
        for (int u = 0; u < 8; ++u) Xt[xr * GPX + xc + u] = xsrc[u];
        __syncthreads();

#pragma unroll
        for (int kk = 0; kk < 4; ++kk) {
            const int kb = kk * 4 + off;
            v2f a;
            a.x = Wt[(wave * 16 + Lm) * GPW + kb];
            a.y = Wt[(wave * 16 + Lm) * GPW + kb + 1];
#pragma unroll
            for (int nt = 0; nt < 8; ++nt) {
                v2f bb;
                bb.x = Xt[kb * GPX + nt * 16 + Lm];
                bb.y = Xt[(kb + 1) * GPX + nt * 16 + Lm];
                acc[nt] = wmma_f32(a, bb, acc[nt]);
            }
        }
    }

#pragma unroll
    for (int nt = 0; nt < 8; ++nt) {
#pragma unroll
        for (int r = 0; r < 8; ++r) {
            const int m = m0 + wave * 16 + r + 8 * hi;
            const int n = n0 + nt * 16 + Lm;
            Yb[(size_t)m * N + n] = acc[nt][r] + bias[m];
        }
    }
}

// -------------------------------------------------------------------------
// Fused banded attention: scores -> mask+rel_pe -> softmax -> AV.
// One workgroup per (query-block of 128, head, batch). 8 waves, 16 rows each.
// Window radius 128 -> per wave the key range is 272 keys = 17 tiles of 16.
// -------------------------------------------------------------------------
#define QPAD 68   // LDS row stride (floats): 68 % 64 == 4 -> conflict-free ops
__global__ __launch_bounds__(256)
void banded_attn(const float* __restrict__ qp, const float* __restrict__ kp,
                 const float* __restrict__ vp,
                 const unsigned char* __restrict__ kv_mask,
                 const float* __restrict__ rel_pe, float* __restrict__ ob) {
    __shared__ float Qs[128 * QPAD];   // Q tile; reused as P staging in AV phase
    __shared__ float KVs[384 * QPAD];  // K window, later overwritten by V window
    __shared__ float pe_s[WIN];
    __shared__ float msk_s[384];       // additive key bias (0 / -1e4 / -inf-ish)

    const int tid = threadIdx.x;
    const int wave = tid >> 5;
    const int lane = tid & 31;
    const int Lm = lane & 15;
    const int hi = lane >> 4;
    const int off = hi * 2;
    const int i0 = blockIdx.x * 128;
    const int h = blockIdx.y;
    const int b = blockIdx.z;

    const float* qh = qp + ((size_t)b * EMBD + h * DH) * TSEQ;
    const float* kh = kp + ((size_t)b * EMBD + h * DH) * TSEQ;
    const float* vh = vp + ((size_t)b * EMBD + h * DH) * TSEQ;

    for (int idx = tid; idx < WIN; idx += 256) pe_s[idx] = rel_pe[h * WIN + idx];
    for (int idx = tid; idx < 384; idx += 256) {
        const int j = i0 - 128 + idx;
        float mb = 0.f;
        if (j < 0 || j >= TSEQ) mb = -3.0e38f;
        else if (!kv_mask[(size_t)b * TSEQ + j]) mb = -10000.f;
        msk_s[idx] = mb;
    }

    // Q tile: (t-row, d) layout from (d, t) global; scale = 1/sqrt(DH)
    {
        const int tt = tid & 127;
        const int db = (tid >> 7) * 32;
#pragma unroll
        for (int u = 0; u < 32; ++u)
            Qs[tt * QPAD + db + u] = qh[(size_t)(db + u) * TSEQ + i0 + tt] * 0.125f;
    }
    // K window rows jr=0..383 (j = i0-128+jr), zero-filled out of range
    for (int rr = tid & 127; rr < 384; rr += 128) {
        const int j = i0 - 128 + rr;
        const int db = (tid >> 7) * 32;
        if (j >= 0 && j < TSEQ) {
#pragma unroll
            for (int u = 0; u < 32; ++u)
                KVs[rr * QPAD + db + u] = kh[(size_t)(db + u) * TSEQ + j];
        } else {
#pragma unroll
            for (int u = 0; u < 32; ++u) KVs[rr * QPAD + db + u] = 0.f;
        }
    }
    __syncthreads();

    const int m0 = wave * 16;  // this wave's query rows (local)
    v8f acc[17];
#pragma unroll
    for (int kt = 0; kt < 17; ++kt) acc[kt] = v8f_zero();

    // S = Q * K^T over d = 64 (16 k-substeps of 4), 17 key tiles
    for (int d4 = 0; d4 < 16; ++d4) {
        const int kb = d4 * 4 + off;
        v2f a;
        a.x = Qs[(m0 + Lm) * QPAD + kb];
        a.y = Qs[(m0 + Lm) * QPAD + kb + 1];
#pragma unroll
        for (int kt = 0; kt < 17; ++kt) {
            v2f bb;
            bb.x = KVs[(m0 + kt * 16 + Lm) * QPAD + kb];
            bb.y = KVs[(m0 + kt * 16 + Lm) * QPAD + kb + 1];
            acc[kt] = wmma_f32(a, bb, acc[kt]);
        }
    }

    // mask + rel_pe + softmax (row = r + 8*hi lives in one 16-lane half)
#pragma unroll
    for (int r = 0; r < 8; ++r) {
        const int row = r + 8 * hi;
        float vals[17];
        float mx = -3.0e38f;
#pragma unroll
        for (int kt = 0; kt < 17; ++kt) {
            const int nloc = kt * 16 + Lm;   // jr - m0
            const int widx = nloc - row;     // window index 0..256 when valid
            const int wc = widx < 0 ? 0 : (widx > 256 ? 256 : widx);
            float sc = acc[kt][r] + pe_s[wc] + msk_s[m0 + nloc];
            sc = (widx >= 0 && widx <= 256) ? sc : -3.0e38f;
            vals[kt] = sc;
            mx = fmaxf(mx, sc);
        }
#pragma unroll
        for (int m = 1; m <= 8; m <<= 1) mx = fmaxf(mx, __shfl_xor(mx, m, 32));
        float sum = 0.f;
#pragma unroll
        for (int kt = 0; kt < 17; ++kt) {
            const float e = __expf(vals[kt] - mx);
            vals[kt] = e;
            sum += e;
        }
#pragma unroll
        for (int m = 1; m <= 8; m <<= 1) sum += __shfl_xor(sum, m, 32);
        const int iq = i0 + m0 + row;
        const float qv = kv_mask[(size_t)b * TSEQ + iq] ? 1.f : 0.f;
        const float inv = qv / sum;
#pragma unroll
        for (int kt = 0; kt < 17; ++kt) acc[kt][r] = vals[kt] * inv;
    }

    __syncthreads();  // all waves done reading K window
    // overwrite K window with V window
    for (int rr = tid & 127; rr < 384; rr += 128) {
        const int j = i0 - 128 + rr;
        const int db = (tid >> 7) * 32;
        if (j >= 0 && j < TSEQ) {
#pragma unroll
            for (int u = 0; u < 32; ++u)
                KVs[rr * QPAD + db + u] = vh[(size_t)(db + u) * TSEQ + j];
        } else {
#pragma unroll
            for (int u = 0; u < 32; ++u) KVs[rr * QPAD + db + u] = 0.f;
        }
    }
    __syncthreads();

    // AV: O(16x64) = P(16x272) x V(272x64), restaging P through own Q stripe
    float* Pl = &Qs[wave * 16 * QPAD];
    v8f oacc[4];
#pragma unroll
    for (int t = 0; t < 4; ++t) oacc[t] = v8f_zero();

#pragma unroll
    for (int chunk = 0; chunk < 4; ++chunk) {
#pragma unroll
        for (int tt = 0; tt < 4; ++tt) {
            const int kt = chunk * 4 + tt;
#pragma unroll
            for (int r = 0; r < 8; ++r)
                Pl[(r + 8 * hi) * QPAD + tt * 16 + Lm] = acc[kt][r];
        }
        asm volatile("s_wait_dscnt 0x0" ::: "memory");
        for (int k4 = 0; k4 < 16; ++k4) {
            const int kb = k4 * 4 + off;
            v2f a;
            a.x = Pl[Lm * QPAD + kb];
            a.y = Pl[Lm * QPAD + kb + 1];
            const int jr0 = m0 + chunk * 64 + k4 * 4;
#pragma unroll
            for (int dt = 0; dt < 4; ++dt) {
                v2f bb;
                bb.x = KVs[(jr0 + off) * QPAD + dt * 16 + Lm];
                bb.y = KVs[(jr0 + off + 1) * QPAD + dt * 16 + Lm];
                oacc[dt] = wmma_f32(a, bb, oacc[dt]);
            }
        }
    }
    // tail tile kt = 16 (keys 256..271)
    {
#pragma unroll
        for (int r = 0; r < 8; ++r)
            Pl[(r + 8 * hi) * QPAD + Lm] = acc[16][r];
        asm volatile("s_wait_dscnt 0x0" ::: "memory");
#pragma unroll
        for (int k4 = 0; k4 < 4; ++k4) {
            const int kb = k4 * 4 + off;
            v2f a;
            a.x = Pl[Lm * QPAD + kb];
            a.y = Pl[Lm * QPAD + kb + 1];
            const int jr0 = m0 + 256 + k4 * 4;
#pragma unroll
            for (int dt = 0; dt < 4; ++dt) {
                v2f bb;
                bb.x = KVs[(jr0 + off) * QPAD + dt * 16 + Lm];
                bb.y = KVs[(jr0 + off + 1) * QPAD + dt * 16 + Lm];
                oacc[dt] = wmma_f32(a, bb, oacc[dt]);
            }
        }
    }

    // store O in (b, c, t) layout so the output GEMM reuses gemm1x1_bias
    float* oh = ob + ((size_t)b * EMBD + h * DH) * TSEQ;
#pragma unroll
    for (int dt = 0; dt < 4; ++dt) {
#pragma unroll
        for (int r = 0; r < 8; ++r) {
            const int row = r + 8 * hi;
            const int iq = i0 + m0 + row;
            const int dd = dt * 16 + Lm;
            oh[(size_t)dd * TSEQ + iq] = oacc[dt][r];
        }
    }
}

extern "C" void kernel_launch(void* const* d_in, const int* in_sizes, int n_in,
                              void* d_out, int out_size, void* d_ws, size_t ws_size,
                              hipStream_t stream) {
    (void)in_sizes; (void)n_in; (void)out_size; (void)ws_size;
    const float* q = (const float*)d_in[0];
    const unsigned char* kvm = (const unsigned char*)d_in[1];
    const float* Wq = (const float*)d_in[2];
    const float* bq = (const float*)d_in[3];
    const float* Wk = (const float*)d_in[4];
    const float* bk = (const float*)d_in[5];
    const float* Wv = (const float*)d_in[6];
    const float* bv = (const float*)d_in[7];
    const float* Wo = (const float*)d_in[8];
    const float* bo = (const float*)d_in[9];
    const float* pe = (const float*)d_in[10];
    float* out = (float*)d_out;

    const size_t nbt = (size_t)BSZ * EMBD * TSEQ;
    float* qp = (float*)d_ws;
    float* kp = qp + nbt;
    float* vp = kp + nbt;
    float* ob = vp + nbt;

    dim3 gg(TSEQ / 128, EMBD / 128, BSZ);
    dim3 bb(256);
    gemm1x1_bias<<<gg, bb, 0, stream>>>(Wq, bq, q, qp, EMBD, EMBD, TSEQ);
    gemm1x1_bias<<<gg, bb, 0, stream>>>(Wk, bk, q, kp, EMBD, EMBD, TSEQ);
    gemm1x1_bias<<<gg, bb, 0, stream>>>(Wv, bv, q, vp, EMBD, EMBD, TSEQ);

    dim3 ga(TSEQ / 128, HEADS, BSZ);
    banded_attn<<<ga, bb, 0, stream>>>(qp, kp, vp, kvm, pe, ob);

    gemm1x1_bias<<<gg, bb, 0, stream>>>(Wo, bo, ob, out, EMBD, EMBD, TSEQ);
}